// GraphNet_74955769249952
// MI455X (gfx1250) — compile-verified
//
#include <hip/hip_runtime.h>

// GraphNet forward, algebraically refactored for MI455X (gfx1250, wave32):
//   tmp  = embed[idx] @ W1                       (WMMA f32 16x16x4, K=256)
//   agg1 = spmm_sum(tmp)  (width 32, fp atomics) ; h = relu(agg1 + b1) in place
//   agg2 = spmm_sum(h)    (width 32, fp atomics)
//   out  = relu([agg2[g1],agg2[g2]] @ Mcat + beff) (WMMA f32 16x16x4, K=64)
// where Mcat = [W2@Wfc_top ; W2@Wfc_bot] (64x256) and beff folds b2/bfc.
//
// B matrices are stored K-pair interleaved: element (k,n) lives at
// ((k>>1)*N + n)*2 + (k&1), so each WMMA B fragment ((k,n),(k+1,n)) is one
// aligned 8-byte load (ds_load_b64 / global_load_b64) with no repacking.

typedef float v2f __attribute__((ext_vector_type(2)));
typedef float v8f __attribute__((ext_vector_type(8)));

#define N_NODES   65536
#define IN_FEATS  256
#define HIDW      32
#define OUT_FEATS 256
#define BATCHN    4096

// ---------------------------------------------------------------- utilities
__global__ __launch_bounds__(256) void zero_kernel(float4* __restrict__ p, int n4) {
  int i = blockIdx.x * 256 + threadIdx.x;
  if (i < n4) { float4 z; z.x = 0.f; z.y = 0.f; z.z = 0.f; z.w = 0.f; p[i] = z; }
}

// Mcat (packed): k<32 -> sum_i W2[k][i]*Wfc[i][j] ; k>=32 -> sum_i W2[k-32][i]*Wfc[256+i][j]
__global__ __launch_bounds__(256) void make_M_kernel(const float* __restrict__ W2,
                                                     const float* __restrict__ Wfc,
                                                     float* __restrict__ Mcat) {
  int t = blockIdx.x * 256 + threadIdx.x;     // 0..16383
  int k = t >> 8;                             // 0..63
  int j = t & 255;
  const float* wrow = W2 + (size_t)(k & 31) * OUT_FEATS;
  const float* wf   = Wfc + ((k < 32) ? 0 : (size_t)OUT_FEATS * OUT_FEATS) + j;
  float s = 0.f;
  #pragma unroll 8
  for (int i = 0; i < OUT_FEATS; ++i) s += wrow[i] * wf[(size_t)i * OUT_FEATS];
  Mcat[(size_t)(((k >> 1) * OUT_FEATS + j) * 2 + (k & 1))] = s;   // K-pair packed
}

__global__ __launch_bounds__(256) void make_beff_kernel(const float* __restrict__ b2,
                                                        const float* __restrict__ Wfc,
                                                        const float* __restrict__ bfc,
                                                        float* __restrict__ beff) {
  int j = threadIdx.x;
  float s = bfc[j];
  #pragma unroll 8
  for (int i = 0; i < OUT_FEATS; ++i)
    s += b2[i] * (Wfc[(size_t)i * OUT_FEATS + j] +
                  Wfc[(size_t)(OUT_FEATS + i) * OUT_FEATS + j]);
  beff[j] = s;
}

// ------------------------------------------- gathered GEMM: tmp = embed[idx] @ W1
// Block = 8 waves, each wave owns a 16-row tile (M) x 32 cols (two 16x16 C frags).
__global__ __launch_bounds__(256) void gather_gemm1_kernel(const int* __restrict__ idx,
                                                           const float* __restrict__ embed,
                                                           const float* __restrict__ W1,
                                                           float* __restrict__ tmp) {
  __shared__ float sW1[IN_FEATS * HIDW];  // 32 KB, K-pair interleaved
  int tid = threadIdx.x;
  for (int i = tid; i < IN_FEATS * HIDW; i += 256) {
    int k = i >> 5, n = i & 31;                         // row-major source (k,n)
    sW1[((k >> 1) * HIDW + n) * 2 + (k & 1)] = W1[i];   // packed dest
  }
  __syncthreads();

  int wave = tid >> 5;
  int lane = tid & 31;
  int half = lane >> 4;      // 0: K pair {0,1}, 1: K pair {2,3} within each 4-step
  int r    = lane & 15;      // A row within tile / B,C column within tile
  int row0 = blockIdx.x * 128 + wave * 16;

  const float* arow = embed + (size_t)idx[row0 + r] * IN_FEATS;

  v8f c0 = {}; v8f c1 = {};
  for (int k0 = 0; k0 < IN_FEATS; k0 += 4) {
    int kp = (k0 >> 1) + half;                           // K-pair index for this lane
    v2f a  = *(const v2f*)&arow[k0 + 2 * half];          // (M=r, K=k0+2h, k0+2h+1)
    v2f b0 = *(const v2f*)&sW1[(kp * HIDW + r) * 2];          // (K..K+1, N=r)
    v2f b1 = *(const v2f*)&sW1[(kp * HIDW + 16 + r) * 2];     // (K..K+1, N=16+r)
    c0 = __builtin_amdgcn_wmma_f32_16x16x4_f32(false, a, false, b0, (short)0, c0, false, false);
    c1 = __builtin_amdgcn_wmma_f32_16x16x4_f32(false, a, false, b1, (short)0, c1, false, false);
  }
  #pragma unroll
  for (int i = 0; i < 8; ++i) {
    int m = row0 + i + 8 * half;                          // C layout: vgpr i -> M=i(+8)
    tmp[(size_t)m * HIDW + r]      = c0[i];
    tmp[(size_t)m * HIDW + 16 + r] = c1[i];
  }
}

// --------------------------------------- SpMM: acc[dst] += feat[src] (width 32)
// One wave per edge, lane = feature channel -> 128 B coalesced load + fp atomic.
__global__ __launch_bounds__(256) void spmm32_kernel(const int* __restrict__ src,
                                                     const int* __restrict__ dst,
                                                     const float* __restrict__ feat,
                                                     float* __restrict__ acc, int nE) {
  int w    = (blockIdx.x * 256 + threadIdx.x) >> 5;
  int lane = threadIdx.x & 31;
  if (w >= nE) return;
  int s = src[w];
  int d = dst[w];
  float v = feat[(size_t)s * HIDW + lane];
  unsafeAtomicAdd(&acc[(size_t)d * HIDW + lane], v);      // global_atomic_add_f32
}

__global__ __launch_bounds__(256) void bias_relu_kernel(float* __restrict__ h,
                                                        const float* __restrict__ b1, int n) {
  int i = blockIdx.x * 256 + threadIdx.x;
  if (i < n) {
    float v = h[i] + b1[i & (HIDW - 1)];
    h[i] = v > 0.f ? v : 0.f;
  }
}

// ------------------- fused tail: out = relu(P @ Mcat + beff), P[b] = agg2 pair rows
// 256 blocks x 16 rows. Wave w covers cols [32w, 32w+32): two C frags, K=64.
__global__ __launch_bounds__(256) void pair_gemm_kernel(const int* __restrict__ g1,
                                                        const int* __restrict__ g2,
                                                        const float* __restrict__ agg2,
                                                        const float* __restrict__ Mcat,
                                                        const float* __restrict__ beff,
                                                        float* __restrict__ out) {
  __shared__ float P[16 * 64];  // 4 KB pair tile, row-major (A frags contiguous)
  int tid  = threadIdx.x;
  int row0 = blockIdx.x * 16;
  for (int i = tid; i < 16 * 64; i += 256) {
    int rr = i >> 6, k = i & 63;
    int node = (k < 32) ? g1[row0 + rr] : g2[row0 + rr];
    P[i] = agg2[(size_t)node * HIDW + (k & 31)];
  }
  __syncthreads();

  int wave = tid >> 5, lane = tid & 31;
  int half = lane >> 4, r = lane & 15;
  int col0 = wave * 32;

  v8f c0 = {}; v8f c1 = {};
  for (int k0 = 0; k0 < 64; k0 += 4) {
    int kp = (k0 >> 1) + half;
    v2f a  = *(const v2f*)&P[r * 64 + k0 + 2 * half];
    v2f b0 = *(const v2f*)&Mcat[(size_t)(kp * OUT_FEATS + col0 + r) * 2];       // packed
    v2f b1 = *(const v2f*)&Mcat[(size_t)(kp * OUT_FEATS + col0 + 16 + r) * 2];  // packed
    c0 = __builtin_amdgcn_wmma_f32_16x16x4_f32(false, a, false, b0, (short)0, c0, false, false);
    c1 = __builtin_amdgcn_wmma_f32_16x16x4_f32(false, a, false, b1, (short)0, c1, false, false);
  }
  #pragma unroll
  for (int i = 0; i < 8; ++i) {
    int m  = row0 + i + 8 * half;
    int n0 = col0 + r, n1 = col0 + 16 + r;
    float v0 = c0[i] + beff[n0];
    float v1 = c1[i] + beff[n1];
    out[(size_t)m * OUT_FEATS + n0] = v0 > 0.f ? v0 : 0.f;
    out[(size_t)m * OUT_FEATS + n1] = v1 > 0.f ? v1 : 0.f;
  }
}

// ------------------------------------------------------------------ launcher
extern "C" void kernel_launch(void* const* d_in, const int* in_sizes, int n_in,
                              void* d_out, int out_size, void* d_ws, size_t ws_size,
                              hipStream_t stream) {
  const int*   idx   = (const int*)d_in[0];
  const int*   src   = (const int*)d_in[1];
  const int*   dst   = (const int*)d_in[2];
  const int*   g1    = (const int*)d_in[3];
  const int*   g2    = (const int*)d_in[4];
  const float* embed = (const float*)d_in[5];
  const float* W1    = (const float*)d_in[6];
  const float* b1    = (const float*)d_in[7];
  const float* W2    = (const float*)d_in[8];
  const float* b2    = (const float*)d_in[9];
  const float* Wfc   = (const float*)d_in[10];
  const float* bfc   = (const float*)d_in[11];
  float* out = (float*)d_out;
  int nE = in_sizes[1];

  char* ws = (char*)d_ws;
  const size_t MB = 1024 * 1024;
  float* tmp  = (float*)(ws);               // [65536,32]  8 MB : embed[idx]@W1
  float* agg1 = (float*)(ws + 8  * MB);     // [65536,32]  8 MB : spmm1 / h (in place)
  float* agg2 = (float*)(ws + 16 * MB);     // [65536,32]  8 MB : spmm2
  float* Mcat = (float*)(ws + 24 * MB);     // [64,256]   64 KB : fused W2@Wfc (packed)
  float* beff = (float*)(ws + 24 * MB + 65536); // [256]        : fused bias

  // zero both accumulators (contiguous region) every call
  zero_kernel<<<4096, 256, 0, stream>>>((float4*)agg1, (2 * N_NODES * HIDW) / 4);
  // tiny fused-weight precompute
  make_M_kernel<<<64, 256, 0, stream>>>(W2, Wfc, Mcat);
  make_beff_kernel<<<1, 256, 0, stream>>>(b2, Wfc, bfc, beff);
  // tmp = embed[idx] @ W1  (WMMA)
  gather_gemm1_kernel<<<N_NODES / 128, 256, 0, stream>>>(idx, embed, W1, tmp);
  // agg1 = spmm(tmp); h = relu(agg1 + b1) in place
  spmm32_kernel<<<nE / 8, 256, 0, stream>>>(src, dst, tmp, agg1, nE);
  bias_relu_kernel<<<(N_NODES * HIDW) / 256, 256, 0, stream>>>(agg1, b1, N_NODES * HIDW);
  // agg2 = spmm(h)
  spmm32_kernel<<<nE / 8, 256, 0, stream>>>(src, dst, agg1, agg2, nE);
  // out = relu(pair @ Mcat + beff)  (WMMA)
  pair_gemm_kernel<<<BATCHN / 16, 256, 0, stream>>>(g1, g2, agg2, Mcat, beff, out);
}